// PedestrianDetector_28415503630416
// MI455X (gfx1250) — compile-verified
//
#include <hip/hip_runtime.h>
#include <math.h>

// PedestrianDetector on MI455X (gfx1250).
// Memory-bound problem: features = 32*512*2048 f32 = 128 MiB streamed once,
// ~5.7us floor at 23.3 TB/s; compute is only ~1 GFLOP. Both linear heads are
// fused into a single N=16 column block (12 bbox + 3 conf + 1 zero pad) and
// executed with V_WMMA_F32_16X16X4_F32 (native fp32 path, no precision loss,
// more than enough throughput for a bandwidth-bound kernel).

typedef __attribute__((ext_vector_type(2))) float v2f;
typedef __attribute__((ext_vector_type(8))) float v8f;

#define WAVES_PER_BLOCK 4
#define NCOL 16   // 12 bbox + 3 conf + 1 pad

// ---------------------------------------------------------------------------
// Pre-pass: pack W_bbox[D,12] and W_conf[D,3] into the per-lane B-matrix
// layout of V_WMMA_F32_16X16X4_F32:
//   B VGPR0 holds K=2*kg  (kg = lane>>4), VGPR1 holds K=2*kg+1, col = lane&15.
// Stored as Wpk[ks*32 + lane] = float2{ W[4*ks+2*kg, col], W[4*ks+2*kg+1, col] }
// so the main loop's B fetch is one fully-coalesced 8B load per lane.
// Total 128 KB -> lives in the 192 MB L2, reused by all 1024 tiles.
// ---------------------------------------------------------------------------
__global__ void pd_pack_w(const float* __restrict__ Wb,
                          const float* __restrict__ Wc,
                          float2* __restrict__ Wpk, int ksteps) {
  int t = blockIdx.x * blockDim.x + threadIdx.x;
  if (t >= ksteps * 32) return;
  int lane = t & 31;
  int ks   = t >> 5;
  int col  = lane & 15;
  int kg   = lane >> 4;
  int k0   = ks * 4 + 2 * kg;
  float x, y;
  if (col < 12) {
    x = Wb[(size_t)k0 * 12 + col];
    y = Wb[(size_t)(k0 + 1) * 12 + col];
  } else if (col < 15) {
    x = Wc[(size_t)k0 * 3 + (col - 12)];
    y = Wc[(size_t)(k0 + 1) * 3 + (col - 12)];
  } else {
    x = 0.f; y = 0.f;            // pad column -> clean zero accumulation
  }
  Wpk[t] = make_float2(x, y);
}

// ---------------------------------------------------------------------------
// Main kernel: one wave32 per 16-token tile (M=16). K loop of D/4 WMMAs.
// A layout (16x4 f32): lane&15 = M row, lane>>4 = K half; VGPR0 = K=2*kg,
// VGPR1 = K=2*kg+1 -> a single 8B load per lane per step.
// Features are read exactly once -> non-temporal loads keep L2 for weights.
// ---------------------------------------------------------------------------
__global__ __launch_bounds__(32 * WAVES_PER_BLOCK)
void pd_main(const float* __restrict__ feat,
             const float* __restrict__ Wpk,
             const float* __restrict__ b_bbox,
             const float* __restrict__ b_conf,
             float* __restrict__ out,
             int tiles, int D, int ksteps, int M) {
  __shared__ float tileC[WAVES_PER_BLOCK][16][NCOL + 1];  // +1: bank-conflict pad

  const int lane = threadIdx.x & 31;
  const int wv   = threadIdx.x >> 5;
  const int tile = blockIdx.x * WAVES_PER_BLOCK + wv;
  const bool active = tile < tiles;            // wave-uniform -> EXEC all-1 inside
  const int ln15 = lane & 15;                  // A row index / C col index
  const int kg   = lane >> 4;

  if (active) {
    const float* arow = feat + (size_t)(tile * 16 + ln15) * D + 2 * kg;
    const v2f* bptr = (const v2f*)Wpk + lane;

    v8f acc = {};
    #pragma unroll 4
    for (int ks = 0; ks < ksteps; ++ks) {
      v2f a = __builtin_nontemporal_load((const v2f*)(arow + 4 * ks)); // streamed once
      v2f b = bptr[(size_t)ks * 32];                                   // L2-hot, coalesced
      // D = A(16x4) * B(4x16) + C ; 8 args: neg_a, A, neg_b, B, c_mod, C, reuse_a, reuse_b
      acc = __builtin_amdgcn_wmma_f32_16x16x4_f32(
          false, a, false, b, (short)0, acc, false, false);
    }

    // C layout: VGPR v -> row v+8*kg, col = lane&15. Scatter to LDS so the
    // epilogue can work token-major.
    #pragma unroll
    for (int v = 0; v < 8; ++v)
      tileC[wv][v + 8 * kg][ln15] = acc[v];
  }

  __syncthreads();

  // Epilogue: lanes 0..15 each own one token of the wave's tile.
  if (active && lane < 16) {
    float bx[12];
    #pragma unroll
    for (int j = 0; j < 12; ++j)
      bx[j] = tileC[wv][lane][j] + b_bbox[j];

    float cf[3];
    #pragma unroll
    for (int a = 0; a < 3; ++a) {
      float z = tileC[wv][lane][12 + a] + b_conf[a];
      cf[a] = 1.f / (1.f + __expf(-z));        // sigmoid
    }

    // Descending sort of 3 (strict '<' compare-exchanges; not stable: if
    // cf0 == cf1 < cf2 the order is 2,1,0, not index-stable 2,0,1).
    int i0 = 0, i1 = 1, i2 = 2, tswap;
    if (cf[i0] < cf[i1]) { tswap = i0; i0 = i1; i1 = tswap; }
    if (cf[i0] < cf[i2]) { tswap = i0; i0 = i2; i2 = tswap; }
    if (cf[i1] < cf[i2]) { tswap = i1; i1 = i2; i2 = tswap; }
    const int ord[3] = { i0, i1, i2 };

    const int token = tile * 16 + lane;
    float* out_boxes = out;                          // [M,3,4]
    float* out_conf  = out + (size_t)M * 12;         // [M,3]
    float* out_valid = out + (size_t)M * 15;         // [M,3] (bool as 0/1 float)

    #pragma unroll
    for (int s = 0; s < 3; ++s) {
      const int idx = ord[s];
      const float c = cf[idx];
      const bool val = c > 0.5f;                     // CONF_THRESH
      out_conf [(size_t)token * 3 + s] = c;
      out_valid[(size_t)token * 3 + s] = val ? 1.f : 0.f;
      #pragma unroll
      for (int j = 0; j < 4; ++j)
        out_boxes[(size_t)token * 12 + s * 4 + j] = val ? bx[idx * 4 + j] : 0.f;
    }
  }
}

// ---------------------------------------------------------------------------
extern "C" void kernel_launch(void* const* d_in, const int* in_sizes, int n_in,
                              void* d_out, int out_size, void* d_ws, size_t ws_size,
                              hipStream_t stream) {
  const float* feat = (const float*)d_in[0];   // [B,T,D] f32
  const float* Wb   = (const float*)d_in[1];   // [D,12]
  const float* bb   = (const float*)d_in[2];   // [12]
  const float* Wc   = (const float*)d_in[3];   // [D,3]
  const float* bc   = (const float*)d_in[4];   // [3]
  float* out = (float*)d_out;

  const int D      = in_sizes[1] / 12;         // 2048
  const int M      = in_sizes[0] / D;          // 16384 tokens
  const int ksteps = D / 4;                    // 512 WMMA steps
  const int tiles  = M / 16;                   // 1024 wave-tiles

  float2* Wpk = (float2*)d_ws;                 // 128 KB packed weights

  {
    int n = ksteps * 32;
    pd_pack_w<<<(n + 255) / 256, 256, 0, stream>>>(Wb, Wc, Wpk, ksteps);
  }
  {
    dim3 block(32 * WAVES_PER_BLOCK);
    dim3 grid((tiles + WAVES_PER_BLOCK - 1) / WAVES_PER_BLOCK);
    pd_main<<<grid, block, 0, stream>>>(feat, (const float*)Wpk, bb, bc, out,
                                        tiles, D, ksteps, M);
  }
}